// CRFCFGMixin_54494545052152
// MI455X (gfx1250) — compile-verified
//
#include <hip/hip_runtime.h>

#define BATCH 8
#define LEN   32
#define HID   1024
#define NT    32
#define NEG10f 1.0e10f
#define NEG15f 1.0e15f

typedef __attribute__((ext_vector_type(2))) float v2f;
typedef __attribute__((ext_vector_type(8))) float v8f;

// ---------------------------------------------------------------------------
// Pre-swizzle a 1024x32 (K x N, row-major) weight matrix into WMMA B-fragment
// layout: out[((kk/4)*2 + coltile)*32 + lane] = {W[kb, n], W[kb+1, n]}
// where half=lane>>4, l15=lane&15, kb = kk + 2*half, n = coltile*16 + l15.
// One v2f per thread, 16384 total.
// ---------------------------------------------------------------------------
__global__ void k_pack_b(const float* __restrict__ W, float* __restrict__ out)
{
  const int gid  = blockIdx.x * 256 + threadIdx.x;  // 16384
  const int lane = gid & 31;
  const int t    = (gid >> 5) & 1;
  const int s    = gid >> 6;                        // k-step (kk = 4s)
  const int half = lane >> 4;
  const int l15  = lane & 15;
  const int kb   = 4 * s + 2 * half;
  const int n    = t * 16 + l15;
  v2f v; v.x = W[(size_t)kb * NT + n]; v.y = W[(size_t)(kb + 1) * NT + n];
  *(v2f*)(out + (size_t)gid * 2) = v;
}

// ---------------------------------------------------------------------------
// Pre-swizzle a 32x1024 (M x K, row-major) matrix into WMMA A-fragment layout:
// out[((kk/4)*2 + rowtile)*32 + lane] = {M[n, kb], M[n, kb+1]}
// where n = rowtile*16 + l15, kb = kk + 2*half.
// ---------------------------------------------------------------------------
__global__ void k_pack_a(const float* __restrict__ M, float* __restrict__ out)
{
  const int gid  = blockIdx.x * 256 + threadIdx.x;  // 16384
  const int lane = gid & 31;
  const int t    = (gid >> 5) & 1;                  // row tile (n0 = 16t)
  const int s    = gid >> 6;
  const int half = lane >> 4;
  const int l15  = lane & 15;
  const int kb   = 4 * s + 2 * half;
  const int n    = t * 16 + l15;
  v2f v; v.x = M[(size_t)n * 1024 + kb]; v.y = M[(size_t)n * 1024 + kb + 1];
  *(v2f*)(out + (size_t)gid * 2) = v;
}

// ---------------------------------------------------------------------------
// Kernel 1: node[b,l,m,:] = phrase[b,l,m,:] @ W_node + b_node   (8192x1024 @ 1024x32)
//           span[b,l,m]   = phrase[b,l,m,:] . W_span + b_span   (fused, VALU)
// One wave per 16-row tile; B fragments come pre-swizzled (coalesced b64 loads).
// ---------------------------------------------------------------------------
__global__ void k_node_span(const float* __restrict__ phrase,
                            const float* __restrict__ WnSwz,
                            const float* __restrict__ bn,
                            const float* __restrict__ Wsp,
                            const float* __restrict__ bsp,
                            float* __restrict__ node,
                            float* __restrict__ span)
{
  const int lane = threadIdx.x & 31;
  const int wave = threadIdx.x >> 5;
  const int tile = blockIdx.x * 4 + wave;          // 512 tiles of 16 rows
  const int half = lane >> 4;                      // 0: K={0,1}, 1: K={2,3}
  const int l15  = lane & 15;
  const int rowA = tile * 16 + l15;
  const float* xrow = phrase + (size_t)rowA * HID + half * 2;
  const v2f*   wsw  = (const v2f*)WnSwz;

  v8f c0 = {}; v8f c1 = {};
  float sp = 0.0f;
  for (int s = 0; s < HID / 4; ++s) {
    v2f a  = *(const v2f*)(xrow + 4 * s);           // A tile (16x4 f32)
    v2f b0 = wsw[(size_t)(s * 2 + 0) * 32 + lane];  // B tiles, coalesced 256B
    v2f b1 = wsw[(size_t)(s * 2 + 1) * 32 + lane];
    c0 = __builtin_amdgcn_wmma_f32_16x16x4_f32(false, a, false, b0, (short)0, c0, false, false);
    c1 = __builtin_amdgcn_wmma_f32_16x16x4_f32(false, a, false, b1, (short)0, c1, false, false);
    v2f wsv = *(const v2f*)(Wsp + 4 * s + half * 2);
    sp += a.x * wsv.x + a.y * wsv.y;                // fused span GEMV
  }
  // lane l and l^16 hold the two K-halves of the same row
  sp += __shfl_xor(sp, 16);
  if (half == 0) span[rowA] = sp + bsp[0];

  const int rowOff = half * 8;                      // C layout: vgpr r -> M=r / M=8+r
  const float bn0 = bn[l15], bn1 = bn[16 + l15];
  #pragma unroll
  for (int r = 0; r < 8; ++r) {
    const int row = tile * 16 + rowOff + r;
    node[(size_t)row * NT + l15]      = c0[r] + bn0;
    node[(size_t)row * NT + 16 + l15] = c1[r] + bn1;
  }
}

// ---------------------------------------------------------------------------
// Kernel 2: posnode = seq_hiddens @ W_posnode + b_posnode + (mask-1)*1e10
// ---------------------------------------------------------------------------
__global__ void k_posnode(const float* __restrict__ seqh,
                          const float* __restrict__ WpSwz,
                          const float* __restrict__ bp,
                          const float* __restrict__ pmask,
                          float* __restrict__ posnode)
{
  const int lane = threadIdx.x & 31;
  const int wave = threadIdx.x >> 5;
  const int tile = blockIdx.x * 4 + wave;          // 16 tiles (256 rows)
  const int half = lane >> 4;
  const int l15  = lane & 15;
  const int rowA = tile * 16 + l15;
  const float* xrow = seqh + (size_t)rowA * HID + half * 2;
  const v2f*   wsw  = (const v2f*)WpSwz;

  v8f c0 = {}; v8f c1 = {};
  for (int s = 0; s < HID / 4; ++s) {
    v2f a  = *(const v2f*)(xrow + 4 * s);
    v2f b0 = wsw[(size_t)(s * 2 + 0) * 32 + lane];
    v2f b1 = wsw[(size_t)(s * 2 + 1) * 32 + lane];
    c0 = __builtin_amdgcn_wmma_f32_16x16x4_f32(false, a, false, b0, (short)0, c0, false, false);
    c1 = __builtin_amdgcn_wmma_f32_16x16x4_f32(false, a, false, b1, (short)0, c1, false, false);
  }
  const int rowOff = half * 8;
  const float bias0 = bp[l15]      + (pmask[l15]      - 1.0f) * NEG10f;
  const float bias1 = bp[16 + l15] + (pmask[16 + l15] - 1.0f) * NEG10f;
  #pragma unroll
  for (int r = 0; r < 8; ++r) {
    const int row = tile * 16 + rowOff + r;
    posnode[(size_t)row * NT + l15]      = c0[r] + bias0;
    posnode[(size_t)row * NT + 16 + l15] = c1[r] + bias1;
  }
}

// ---------------------------------------------------------------------------
// Kernel 3a: Er[n,ac] = exp(rule[n,ac] + (mask-1)*1e10 - Mr[n]); Mr[n] = max_ac
// ---------------------------------------------------------------------------
__global__ void k_rule_exp(const float* __restrict__ rule,
                           const float* __restrict__ rmask,
                           float* __restrict__ Er,
                           float* __restrict__ Mr)
{
  const int n   = blockIdx.x;
  const int tid = threadIdx.x;                     // 256
  __shared__ float red[256];
  float v[4]; float lm = -INFINITY;
  #pragma unroll
  for (int q = 0; q < 4; ++q) {
    const int e = tid + q * 256;
    const float x = rule[n * 1024 + e] + (rmask[n * 1024 + e] - 1.0f) * NEG10f;
    v[q] = x; lm = fmaxf(lm, x);
  }
  red[tid] = lm; __syncthreads();
  for (int off = 128; off; off >>= 1) {
    if (tid < off) red[tid] = fmaxf(red[tid], red[tid + off]);
    __syncthreads();
  }
  const float m = red[0];
  if (tid == 0) Mr[n] = m;
  #pragma unroll
  for (int q = 0; q < 4; ++q) {
    const int e = tid + q * 256;
    Er[n * 1024 + e] = __expf(v[q] - m);
  }
}

// ---------------------------------------------------------------------------
// Kernel 3b: chart[b,l,l,n] = node[b,l,l,n] + lse_p( pos_unary'[n,p] + posnode[b,l,p] )
// ---------------------------------------------------------------------------
__global__ void k_diag(const float* __restrict__ posnode,
                       const float* __restrict__ pu,
                       const float* __restrict__ pum,
                       const float* __restrict__ node,
                       float* __restrict__ chart)
{
  const int bl = blockIdx.x;                       // b*L + l
  const int l  = bl & (LEN - 1);
  const int n  = threadIdx.x;                      // 32
  __shared__ float pn[NT];
  pn[n] = posnode[bl * NT + n];
  __syncthreads();
  float m = -INFINITY, s = 0.0f;
  for (int p = 0; p < NT; ++p) {
    const float v = pu[n * NT + p] + (pum[n * NT + p] - 1.0f) * NEG15f + pn[p];
    if (v > m) { s = s * __expf(m - v) + 1.0f; m = v; }
    else       { s += __expf(v - m); }
  }
  const size_t idx = ((size_t)bl * LEN + l) * NT + n;
  chart[idx] = node[idx] + m + __logf(s);
}

// ---------------------------------------------------------------------------
// Kernel 4a (per width i): for each span (b,t): s[a,c] = lse_j(left_j[a]+right_j[c]),
// Ms = max s, Es[span,ac] = exp(s - Ms).  LDS-staged, online lse.
// ---------------------------------------------------------------------------
__global__ void k_spans_es(const float* __restrict__ chart,
                           const float* __restrict__ node,
                           float* __restrict__ Es,
                           float* __restrict__ Msb,
                           int i)
{
  const int nspan = LEN - i;
  const int span  = blockIdx.x;                    // b*(L-i) + t
  const int b = span / nspan;
  const int t = span - b * nspan;
  const int tid = threadIdx.y * 32 + threadIdx.x;  // block (32,8)
  __shared__ float Lsh[(LEN - 1) * NT];
  __shared__ float Rsh[(LEN - 1) * NT];
  __shared__ float red[9];

  const int total = i * NT;
  for (int e = tid; e < total; e += 256) {
    const int j = e >> 5, a = e & 31;
    const size_t li = ((size_t)(b * LEN + t) * LEN + (t + j)) * NT + a;
    Lsh[e] = chart[li] + node[li];
    const size_t ri = ((size_t)(b * LEN + (t + j + 1)) * LEN + (t + i)) * NT + a;
    Rsh[e] = chart[ri] + node[ri];
  }
  __syncthreads();

  float sv[4]; float lm = -INFINITY;
  #pragma unroll
  for (int q = 0; q < 4; ++q) {
    const int ac = tid + q * 256;
    const int a = ac >> 5, cc = ac & 31;
    float m = -INFINITY, ss = 0.0f;
    for (int j = 0; j < i; ++j) {
      const float v = Lsh[j * NT + a] + Rsh[j * NT + cc];
      if (v > m) { ss = ss * __expf(m - v) + 1.0f; m = v; }
      else       { ss += __expf(v - m); }
    }
    const float s = m + __logf(ss);
    sv[q] = s; lm = fmaxf(lm, s);
  }
  for (int off = 16; off; off >>= 1) lm = fmaxf(lm, __shfl_xor(lm, off));
  if (threadIdx.x == 0) red[threadIdx.y] = lm;
  __syncthreads();
  if (tid == 0) {
    float m = red[0];
    for (int w = 1; w < 8; ++w) m = fmaxf(m, red[w]);
    red[8] = m; Msb[span] = m;
  }
  __syncthreads();
  const float Ms = red[8];
  float* eo = Es + (size_t)span * 1024;
  #pragma unroll
  for (int q = 0; q < 4; ++q) {
    const int ac = tid + q * 256;
    eo[ac] = __expf(sv[q] - Ms);
  }
}

// ---------------------------------------------------------------------------
// Kernel 4b (per width i): chart[b,t,t+i,n] = log( Er(32x1024) @ Es(1024xspans) )
//   + Mr[n] + Ms[span] + node[b,t,t+i,n] + span[b,t,t+i]      — WMMA GEMM.
// A fragments (Er) come pre-swizzled; one wave per 16x16 output tile.
// ---------------------------------------------------------------------------
__global__ void k_rule_wmma(const float* __restrict__ ErSwz,
                            const float* __restrict__ Mr,
                            const float* __restrict__ Es,
                            const float* __restrict__ Msb,
                            const float* __restrict__ node,
                            const float* __restrict__ spansc,
                            float* __restrict__ chart,
                            int i)
{
  const int nspan = LEN - i;
  const int spans = BATCH * nspan;
  const int lane  = threadIdx.x;
  const int half  = lane >> 4;
  const int l15   = lane & 15;
  const int rt    = blockIdx.y;                    // row tile: n0 = 16*rt
  const int col   = blockIdx.x * 16 + l15;
  const bool valid = col < spans;

  const v2f*   asw  = (const v2f*)ErSwz;
  const float* brow = Es + (valid ? (size_t)col * 1024 : 0) + half * 2;

  v8f c = {};
  for (int s = 0; s < 256; ++s) {
    v2f a = asw[(size_t)(s * 2 + rt) * 32 + lane]; // coalesced 256B A fragment
    v2f bv;
    if (valid) { bv = *(const v2f*)(brow + 4 * s); }
    else       { bv.x = 0.0f; bv.y = 0.0f; }
    c = __builtin_amdgcn_wmma_f32_16x16x4_f32(false, a, false, bv, (short)0, c, false, false);
  }
  if (!valid) return;

  const int b = col / nspan;
  const int t = col - b * nspan;
  const size_t sidx  = (size_t)(b * LEN + t) * LEN + (t + i);
  const size_t rbase = sidx * NT;
  const float add = Msb[col] + spansc[sidx];
  const int nb = rt * 16 + half * 8;
  float out[8];
  #pragma unroll
  for (int r = 0; r < 8; ++r)
    out[r] = __logf(c[r]) + Mr[nb + r] + add + node[rbase + nb + r];
  float4 o0 = {out[0], out[1], out[2], out[3]};
  float4 o1 = {out[4], out[5], out[6], out[7]};
  *(float4*)(chart + rbase + nb)     = o0;
  *(float4*)(chart + rbase + nb + 4) = o1;
}

// ---------------------------------------------------------------------------
// Kernel 5: logits[b,n] = chart[b,0,seq_len-1,n] + (root_mask[n]-1)*1e10
// ---------------------------------------------------------------------------
__global__ void k_logits(const float* __restrict__ chart,
                         const int* __restrict__ seq_masks,
                         const float* __restrict__ root_mask,
                         float* __restrict__ out)
{
  const int b = blockIdx.x;
  const int n = threadIdx.x;
  int sl = 0;
  for (int l = 0; l < LEN; ++l) sl += seq_masks[b * LEN + l];
  const size_t idx = ((size_t)(b * LEN) * LEN + (sl - 1)) * NT + n;
  out[b * NT + n] = chart[idx] + (root_mask[n] - 1.0f) * NEG10f;
}

// ---------------------------------------------------------------------------
extern "C" void kernel_launch(void* const* d_in, const int* in_sizes, int n_in,
                              void* d_out, int out_size, void* d_ws, size_t ws_size,
                              hipStream_t stream) {
  (void)in_sizes; (void)n_in; (void)out_size; (void)ws_size;
  const float* phrase = (const float*)d_in[0];
  const float* seqh   = (const float*)d_in[1];
  const int*   smask  = (const int*)  d_in[2];
  const float* Wp     = (const float*)d_in[3];
  const float* bp     = (const float*)d_in[4];
  const float* Wn     = (const float*)d_in[5];
  const float* bn     = (const float*)d_in[6];
  const float* Wsp    = (const float*)d_in[7];
  const float* bsp    = (const float*)d_in[8];
  const float* rule   = (const float*)d_in[9];
  const float* pu     = (const float*)d_in[10];
  const float* rootm  = (const float*)d_in[11];
  const float* pnm    = (const float*)d_in[12];
  const float* rulem  = (const float*)d_in[13];
  const float* pum    = (const float*)d_in[14];

  // workspace partitions (floats), all 256B-aligned
  float* ws    = (float*)d_ws;
  float* node  = ws;                 // 262144  (B,L,L,N)
  float* span  = node + 262144;      // 8192    (B,L,L)
  float* posn  = span + 8192;        // 8192    (B,L,N)
  float* chart = posn + 8192;        // 262144  (B,L,L,N)
  float* Er    = chart + 262144;     // 32768   (N, N*N)
  float* Mr    = Er + 32768;         // 64 (32 used)
  float* Es    = Mr + 64;            // 253952  (maxspans=248, 1024)
  float* Msb   = Es + 253952;        // 256
  float* WnSwz = Msb + 256;          // 32768   (B-fragment layout)
  float* WpSwz = WnSwz + 32768;      // 32768
  float* ErSwz = WpSwz + 32768;      // 32768   (A-fragment layout)

  k_pack_b<<<64, 256, 0, stream>>>(Wn, WnSwz);
  k_pack_b<<<64, 256, 0, stream>>>(Wp, WpSwz);
  k_node_span<<<128, 128, 0, stream>>>(phrase, WnSwz, bn, Wsp, bsp, node, span);
  k_posnode  <<<4,   128, 0, stream>>>(seqh, WpSwz, bp, pnm, posn);
  k_rule_exp <<<32,  256, 0, stream>>>(rule, rulem, Er, Mr);
  k_pack_a   <<<64,  256, 0, stream>>>(Er, ErSwz);
  k_diag     <<<BATCH * LEN, 32, 0, stream>>>(posn, pu, pum, node, chart);

  for (int i = 1; i < LEN; ++i) {
    const int spans = BATCH * (LEN - i);
    k_spans_es<<<spans, dim3(32, 8), 0, stream>>>(chart, node, Es, Msb, i);
    dim3 g((spans + 15) / 16, 2);
    k_rule_wmma<<<g, 32, 0, stream>>>(ErSwz, Mr, Es, Msb, node, span, chart, i);
  }
  k_logits<<<BATCH, 32, 0, stream>>>(chart, smask, rootm, (float*)d_out);
}